// GPT_89180700934424
// MI455X (gfx1250) — compile-verified
//
#include <hip/hip_runtime.h>
#include <hip/hip_bf16.h>
#include <math.h>

#define V_  16384
#define D_  512
#define H_  2048
#define L_  2
#define C_  8192
#define W_  256
#define CB_ 32

typedef float v2f __attribute__((ext_vector_type(2)));
typedef float v8f __attribute__((ext_vector_type(8)));

// ---------------------------------------------------------------------------
// Elementwise / reduction helpers
// ---------------------------------------------------------------------------

__global__ __launch_bounds__(256) void embed_kernel(const int* __restrict__ x,
        const float* __restrict__ cte, const float* __restrict__ cpe,
        float* __restrict__ h) {
    int idx = blockIdx.x * 256 + threadIdx.x;      // over C_*D_
    int row = idx >> 9;                            // / D_
    int d   = idx & (D_ - 1);
    h[idx] = cte[(size_t)x[row] * D_ + d] + cpe[idx];
}

// LayerNorm over rows of length D_=512; 256 threads, 2 elems/thread.
// Values cached in registers -> safe for in-place (out == in).
__global__ __launch_bounds__(256) void ln_kernel(const float* __restrict__ in,
        float* __restrict__ out, const float* __restrict__ g,
        const float* __restrict__ b) {
    __shared__ float red[256];
    int row = blockIdx.x;
    const float* xr = in + (size_t)row * D_;
    float vals[2];
    float s = 0.f;
#pragma unroll
    for (int i = 0; i < 2; ++i) { vals[i] = xr[threadIdx.x + i * 256]; s += vals[i]; }
    red[threadIdx.x] = s; __syncthreads();
    for (int off = 128; off > 0; off >>= 1) {
        if (threadIdx.x < off) red[threadIdx.x] += red[threadIdx.x + off];
        __syncthreads();
    }
    float mean = red[0] * (1.0f / D_);
    __syncthreads();
    float s2 = 0.f;
#pragma unroll
    for (int i = 0; i < 2; ++i) { float dv = vals[i] - mean; s2 += dv * dv; }
    red[threadIdx.x] = s2; __syncthreads();
    for (int off = 128; off > 0; off >>= 1) {
        if (threadIdx.x < off) red[threadIdx.x] += red[threadIdx.x + off];
        __syncthreads();
    }
    float rstd = rsqrtf(red[0] * (1.0f / D_) + 1e-5f);
    float* orow = out + (size_t)row * D_;
#pragma unroll
    for (int i = 0; i < 2; ++i) {
        int d = threadIdx.x + i * 256;
        orow[d] = (vals[i] - mean) * rstd * g[d] + b[d];
    }
}

__global__ __launch_bounds__(256) void gather_rows(const float* __restrict__ src,
        const int* __restrict__ am, float* __restrict__ dst) {
    int idx = blockIdx.x * 256 + threadIdx.x;      // over W_*D_
    int w = idx >> 9; int d = idx & (D_ - 1);
    dst[idx] = src[(size_t)am[w] * D_ + d];
}

__global__ __launch_bounds__(256) void scatter_add_rows(float* __restrict__ dst,
        const int* __restrict__ am, const float* __restrict__ src) {
    int idx = blockIdx.x * 256 + threadIdx.x;      // over W_*D_
    int w = idx >> 9; int d = idx & (D_ - 1);
    dst[(size_t)am[w] * D_ + d] += src[idx];       // rows unique -> no atomics
}

// Masked softmax over row w of sc (length C_): allowed prefix = am[w]+1,
// probabilities beyond the prefix are exactly 0.
__global__ __launch_bounds__(256) void softmax_mask_kernel(float* __restrict__ sc,
        const int* __restrict__ am) {
    __shared__ float red[256];
    int w = blockIdx.x;
    float* row = sc + (size_t)w * C_;
    int limit = am[w] + 1;
    float mx = -1e30f;
    for (int c = threadIdx.x; c < limit; c += 256) mx = fmaxf(mx, row[c]);
    red[threadIdx.x] = mx; __syncthreads();
    for (int off = 128; off > 0; off >>= 1) {
        if (threadIdx.x < off) red[threadIdx.x] = fmaxf(red[threadIdx.x], red[threadIdx.x + off]);
        __syncthreads();
    }
    mx = red[0]; __syncthreads();
    float s = 0.f;
    for (int c = threadIdx.x; c < limit; c += 256) {
        float e = expf(row[c] - mx); row[c] = e; s += e;
    }
    red[threadIdx.x] = s; __syncthreads();
    for (int off = 128; off > 0; off >>= 1) {
        if (threadIdx.x < off) red[threadIdx.x] += red[threadIdx.x + off];
        __syncthreads();
    }
    float inv = 1.0f / red[0];
    for (int c = threadIdx.x; c < limit; c += 256) row[c] *= inv;
    for (int c = limit + threadIdx.x; c < C_; c += 256) row[c] = 0.f;
}

__global__ __launch_bounds__(256) void gelu_kernel(float* __restrict__ a, int n) {
    int idx = blockIdx.x * 256 + threadIdx.x;
    if (idx < n) {
        float xv = a[idx];
        a[idx] = 0.5f * xv * (1.0f + erff(xv * 0.70710678118654752f));
    }
}

__global__ __launch_bounds__(256) void addpe_kernel(float* __restrict__ z,
        const float* __restrict__ cpe, const float* __restrict__ wpe) {
    int idx = blockIdx.x * 256 + threadIdx.x;      // over W_*CB_*D_
    int d = idx & (D_ - 1);
    int c = (idx >> 9) & (CB_ - 1);
    int w = idx >> 14;                             // / (CB_*D_)
    z[idx] += cpe[c * D_ + d] + wpe[w * D_ + d];
}

// ---------------------------------------------------------------------------
// Generic fp32 WMMA GEMM with 2x2 register blocking per wave:
//   C[m,n] = sum_k A[m,k] * Bk(k,n)   (+bias[n]) (+resid[m,n])
//   BT=true : B is (N,K) row-major, op = A @ B^T   (contiguous K both sides)
//   BT=false: B is (K,N) row-major, op = A @ B
// Each wave owns a 32(M) x 32(N) macro-tile = 2x2 grid of 16x16 WMMA tiles:
// 4 x v_wmma_f32_16x16x4_f32 per 4 operand loads -> 8 FLOP/byte from cache.
// Block = 128 threads = 4 waves side-by-side in N: block tile 32(M) x 128(N).
// Grid = (N/128, M/32).  All dims divide exactly; EXEC all-ones everywhere.
// Fragment layouts per CDNA5 ISA 7.12.2:
//   A/B frag[i] = Mtx[lane&15][k0 + 2*(lane>>4) + i]
//   C/D acc[r]  = C[mtile + 8*(lane>>4) + r][ntile + (lane&15)]
// ---------------------------------------------------------------------------
template<bool BT>
__global__ __launch_bounds__(128) void gemm_wmma_kernel(
        const float* __restrict__ A, const float* __restrict__ B,
        float* __restrict__ Cm,
        const float* __restrict__ bias, const float* __restrict__ resid,
        int K, int lda, int ldb, int ldc) {
    int wave = threadIdx.x >> 5;
    int lane = threadIdx.x & 31;
    int hi = lane >> 4;
    int lo = lane & 15;
    int m0 = blockIdx.y << 5;                      // 32-row macro tile
    int n0 = (blockIdx.x << 7) + (wave << 5);      // 32-col macro tile

    const float* arow0 = A + (size_t)(m0 + lo) * lda + (hi << 1);
    const float* arow1 = arow0 + (size_t)16 * lda;
    v8f acc00 = {}, acc01 = {}, acc10 = {}, acc11 = {};

    if (BT) {
        const float* brow0 = B + (size_t)(n0 + lo) * ldb + (hi << 1);
        const float* brow1 = brow0 + (size_t)16 * ldb;
        for (int k0 = 0; k0 < K; k0 += 4) {
            v2f a0 = *reinterpret_cast<const v2f*>(arow0 + k0);
            v2f a1 = *reinterpret_cast<const v2f*>(arow1 + k0);
            v2f b0 = *reinterpret_cast<const v2f*>(brow0 + k0);
            v2f b1 = *reinterpret_cast<const v2f*>(brow1 + k0);
            acc00 = __builtin_amdgcn_wmma_f32_16x16x4_f32(false, a0, false, b0, (short)0, acc00, false, false);
            acc01 = __builtin_amdgcn_wmma_f32_16x16x4_f32(false, a0, false, b1, (short)0, acc01, false, false);
            acc10 = __builtin_amdgcn_wmma_f32_16x16x4_f32(false, a1, false, b0, (short)0, acc10, false, false);
            acc11 = __builtin_amdgcn_wmma_f32_16x16x4_f32(false, a1, false, b1, (short)0, acc11, false, false);
        }
    } else {
        const float* bcol0 = B + (n0 + lo) + (size_t)(hi << 1) * ldb;
        const float* bcol1 = bcol0 + 16;
        for (int k0 = 0; k0 < K; k0 += 4) {
            v2f a0 = *reinterpret_cast<const v2f*>(arow0 + k0);
            v2f a1 = *reinterpret_cast<const v2f*>(arow1 + k0);
            v2f b0, b1;
            b0.x = bcol0[(size_t)k0 * ldb];
            b0.y = bcol0[(size_t)(k0 + 1) * ldb];
            b1.x = bcol1[(size_t)k0 * ldb];
            b1.y = bcol1[(size_t)(k0 + 1) * ldb];
            acc00 = __builtin_amdgcn_wmma_f32_16x16x4_f32(false, a0, false, b0, (short)0, acc00, false, false);
            acc01 = __builtin_amdgcn_wmma_f32_16x16x4_f32(false, a0, false, b1, (short)0, acc01, false, false);
            acc10 = __builtin_amdgcn_wmma_f32_16x16x4_f32(false, a1, false, b0, (short)0, acc10, false, false);
            acc11 = __builtin_amdgcn_wmma_f32_16x16x4_f32(false, a1, false, b1, (short)0, acc11, false, false);
        }
    }

    // Epilogue: store the 2x2 grid of 16x16 tiles.
    v8f accs[2][2] = { { acc00, acc01 }, { acc10, acc11 } };
#pragma unroll
    for (int mt = 0; mt < 2; ++mt) {
#pragma unroll
        for (int nt = 0; nt < 2; ++nt) {
            int n = n0 + (nt << 4) + lo;
            float badd = bias ? bias[n] : 0.f;
            v8f acc = accs[mt][nt];
#pragma unroll
            for (int r = 0; r < 8; ++r) {
                int m = m0 + (mt << 4) + (hi << 3) + r;
                float v = acc[r] + badd;
                if (resid) v += resid[(size_t)m * ldc + n];
                Cm[(size_t)m * ldc + n] = v;
            }
        }
    }
}

// ---------------------------------------------------------------------------
// Launcher
// ---------------------------------------------------------------------------
extern "C" void kernel_launch(void* const* d_in, const int* in_sizes, int n_in,
                              void* d_out, int out_size, void* d_ws, size_t ws_size,
                              hipStream_t stream) {
    (void)in_sizes; (void)n_in; (void)out_size; (void)ws_size;

    const int*   x       = (const int*)d_in[0];
    const int*   am      = (const int*)d_in[1];
    const float* cte     = (const float*)d_in[2];
    const float* cpe     = (const float*)d_in[3];
    const float* wpe     = (const float*)d_in[4];
    const float* q_w     = (const float*)d_in[5];
    const float* kv_w    = (const float*)d_in[6];
    const float* aproj_w = (const float*)d_in[7];
    const float* fc_w    = (const float*)d_in[8];
    const float* cproj_w = (const float*)d_in[9];
    const float* ln1_g   = (const float*)d_in[10];
    const float* ln1_b   = (const float*)d_in[11];
    const float* ln2_g   = (const float*)d_in[12];
    const float* ln2_b   = (const float*)d_in[13];
    const float* ln3_g   = (const float*)d_in[14];
    const float* ln3_b   = (const float*)d_in[15];
    const float* fproj_w = (const float*)d_in[16];
    const float* fproj_b = (const float*)d_in[17];
    const float* fln1_g  = (const float*)d_in[18];
    const float* fln1_b  = (const float*)d_in[19];
    const float* fln2_g  = (const float*)d_in[20];
    const float* fln2_b  = (const float*)d_in[21];

    float* ws = (float*)d_ws;
    size_t off = 0;
    auto alloc = [&](size_t nelem) { float* p = ws + off; off += nelem; return p; };
    float* h     = alloc((size_t)C_ * D_);
    float* ln1x  = alloc((size_t)C_ * D_);
    float* xbuf  = alloc((size_t)C_ * D_);
    float* kv    = alloc((size_t)C_ * 2 * D_);
    float* q     = alloc((size_t)W_ * D_);
    float* sc    = alloc((size_t)W_ * C_);
    float* aout  = alloc((size_t)W_ * D_);
    float* w_in  = alloc((size_t)W_ * D_);
    float* w2    = alloc((size_t)W_ * D_);
    float* in_w  = alloc((size_t)W_ * D_);
    float* fcact = alloc((size_t)W_ * H_);
    float* words = alloc((size_t)W_ * D_);
    float* wln   = alloc((size_t)W_ * D_);
    float* z     = alloc((size_t)W_ * CB_ * D_);
    float* logits = (float*)d_out;

    embed_kernel<<<(C_ * D_) / 256, 256, 0, stream>>>(x, cte, cpe, h);

    for (int l = 0; l < L_; ++l) {
        const float* qw  = q_w     + (size_t)l * D_ * D_;
        const float* kvw = kv_w    + (size_t)l * 2 * D_ * D_;
        const float* apw = aproj_w + (size_t)l * D_ * D_;
        const float* fcw = fc_w    + (size_t)l * H_ * D_;
        const float* cpw = cproj_w + (size_t)l * D_ * H_;

        ln_kernel<<<C_, 256, 0, stream>>>(h, ln1x, ln1_g + l * D_, ln1_b + l * D_);
        gather_rows<<<(W_ * D_) / 256, 256, 0, stream>>>(ln1x, am, w_in);
        // q = w_in @ qw^T                    (M=W, N=D, K=D)
        gemm_wmma_kernel<true><<<dim3(D_ / 128, W_ / 32), 128, 0, stream>>>(
            w_in, qw, q, nullptr, nullptr, D_, D_, D_, D_);
        // kv = ln1x @ kvw^T                  (M=C, N=2D, K=D)
        gemm_wmma_kernel<true><<<dim3((2 * D_) / 128, C_ / 32), 128, 0, stream>>>(
            ln1x, kvw, kv, nullptr, nullptr, D_, D_, D_, 2 * D_);
        // sc = q @ k^T   (k = kv[:, :D], ldb = 2D)   (M=W, N=C, K=D)
        gemm_wmma_kernel<true><<<dim3(C_ / 128, W_ / 32), 128, 0, stream>>>(
            q, kv, sc, nullptr, nullptr, D_, D_, 2 * D_, C_);
        softmax_mask_kernel<<<W_, 256, 0, stream>>>(sc, am);
        // aout = P @ v   (v = kv[:, D:], NN, ldb = 2D)   (M=W, N=D, K=C)
        gemm_wmma_kernel<false><<<dim3(D_ / 128, W_ / 32), 128, 0, stream>>>(
            sc, kv + D_, aout, nullptr, nullptr, C_, C_, 2 * D_, D_);
        // xbuf = ln1x;  xbuf[am] += aout
        hipMemcpyAsync(xbuf, ln1x, (size_t)C_ * D_ * sizeof(float),
                       hipMemcpyDeviceToDevice, stream);
        scatter_add_rows<<<(W_ * D_) / 256, 256, 0, stream>>>(xbuf, am, aout);
        // w2 = aout @ apw^T + w_in           (M=W, N=D, K=D)
        gemm_wmma_kernel<true><<<dim3(D_ / 128, W_ / 32), 128, 0, stream>>>(
            aout, apw, w2, nullptr, w_in, D_, D_, D_, D_);
        ln_kernel<<<W_, 256, 0, stream>>>(w2, in_w, ln2_g + l * D_, ln2_b + l * D_);
        // h = LN(xbuf) with ln3 (written directly into h)
        ln_kernel<<<C_, 256, 0, stream>>>(xbuf, h, ln3_g + l * D_, ln3_b + l * D_);
        // fcact = in_w @ fcw^T               (M=W, N=H, K=D)
        gemm_wmma_kernel<true><<<dim3(H_ / 128, W_ / 32), 128, 0, stream>>>(
            in_w, fcw, fcact, nullptr, nullptr, D_, D_, D_, H_);
        gelu_kernel<<<(W_ * H_) / 256, 256, 0, stream>>>(fcact, W_ * H_);
        // words = gelu(fcact) @ cpw^T + in_w (M=W, N=D, K=H)
        gemm_wmma_kernel<true><<<dim3(D_ / 128, W_ / 32), 128, 0, stream>>>(
            fcact, cpw, words, nullptr, in_w, H_, H_, H_, D_);
        scatter_add_rows<<<(W_ * D_) / 256, 256, 0, stream>>>(h, am, words);
    }

    ln_kernel<<<W_, 256, 0, stream>>>(words, wln, fln1_g, fln1_b);
    // z = wln @ fproj^T + fproj_b            (M=W, N=CB*D, K=D)
    gemm_wmma_kernel<true><<<dim3((CB_ * D_) / 128, W_ / 32), 128, 0, stream>>>(
        wln, fproj_w, z, fproj_b, nullptr, D_, D_, D_, CB_ * D_);
    addpe_kernel<<<(W_ * CB_ * D_) / 256, 256, 0, stream>>>(z, cpe, wpe);
    ln_kernel<<<W_ * CB_, 256, 0, stream>>>(z, z, fln2_g, fln2_b);   // in-place
    // logits = z @ cte^T                     (M=W*CB=8192, N=V=16384, K=D)
    gemm_wmma_kernel<true><<<dim3(V_ / 128, (W_ * CB_) / 32), 128, 0, stream>>>(
        z, cte, logits, nullptr, nullptr, D_, D_, D_, V_);
}